// Encoder_Postnet_67482526155451
// MI455X (gfx1250) — compile-verified
//
#include <hip/hip_runtime.h>

typedef float v2f __attribute__((ext_vector_type(2)));
typedef float v8f __attribute__((ext_vector_type(8)));

#define Bn 32
#define Tn 4096
#define Pn 512
#define Hn 512

// ---------------------------------------------------------------------------
// Kernel 1: length-regulator index scan.
// idx[b,t] = clip(cumsum_t(align[b,t] != align[b,t-1]), 0, P-1)
// One block per batch; 1024 threads x 4 elements; LDS Hillis-Steele scan.
// ---------------------------------------------------------------------------
__global__ __launch_bounds__(1024) void align_scan_kernel(
    const int* __restrict__ align, int* __restrict__ idx) {
  __shared__ int sums[1024];
  const int b = blockIdx.x;
  const int* ap = align + (size_t)b * Tn;
  int* ip = idx + (size_t)b * Tn;
  const int tid = threadIdx.x;
  const int base = tid * 4;

  int prev = (base > 0) ? ap[base - 1] : ap[0];
  int cnt[4];
  int local = 0;
#pragma unroll
  for (int i = 0; i < 4; ++i) {
    const int cur = ap[base + i];
    if ((base + i) > 0 && cur != prev) ++local;
    cnt[i] = local;
    prev = cur;
  }
  sums[tid] = local;
  __syncthreads();
  // inclusive block scan of per-thread sums
  for (int off = 1; off < 1024; off <<= 1) {
    const int add = (tid >= off) ? sums[tid - off] : 0;
    __syncthreads();
    sums[tid] += add;
    __syncthreads();
  }
  const int excl = sums[tid] - local;
#pragma unroll
  for (int i = 0; i < 4; ++i) {
    int v = excl + cnt[i];
    ip[base + i] = (v < (Pn - 1)) ? v : (Pn - 1);
  }
}

// ---------------------------------------------------------------------------
// Kernel 2: fused gather + 3 affine broadcasts as D = A(16x4) x B(4x16) + C
// using V_WMMA_F32_16X16X4_F32. One wave per 16t x 16h tile; 8 waves/block
// cover 16t x 128h. Bandwidth-bound: ~512MB total traffic @ 23.3 TB/s.
//
// A layout (16x4 f32, 2 VGPRs/lane): M = lane%16; lanes 0-15 hold K=0,1;
//   lanes 16-31 hold K=2,3.
// B layout (4x16 f32, 2 VGPRs/lane): N = lane%16; lanes 0-15 hold K=0,1;
//   lanes 16-31 hold K=2,3.
// C/D layout (16x16 f32, 8 VGPRs): N = lane%16; VGPR r -> M = r + 8*(lane>=16).
// ---------------------------------------------------------------------------
__global__ __launch_bounds__(256) void postnet_wmma_kernel(
    const float* __restrict__ enc, const float* __restrict__ pitch,
    const float* __restrict__ beats,
    const float* __restrict__ w_pitch, const float* __restrict__ b_pitch,
    const float* __restrict__ w_beats, const float* __restrict__ b_beats,
    const float* __restrict__ w_pos, const float* __restrict__ b_pos,
    const int* __restrict__ idx, float* __restrict__ out) {
  const int blk  = blockIdx.x;
  const int hg   = blk & 3;            // 4 groups of 128 channels
  const int tt   = (blk >> 2) & 255;   // 256 time tiles of 16
  const int b    = blk >> 10;          // batch
  const int wave = threadIdx.x >> 5;
  const int lane = threadIdx.x & 31;
  const int h0   = hg * 128 + wave * 16;
  const int t0   = tt * 16;
  const int m    = lane & 15;   // M for A, N for B/C/D
  const int hi   = lane >> 4;   // half-wave selector

  // A operand: rows [pitch, beats, pos, 1]
  const int tA = t0 + m;
  v2f a;
  if (!hi) {
    a.x = pitch[(size_t)b * Tn + tA];
    a.y = beats[(size_t)b * Tn + tA];
  } else {
    a.x = (float)tA;
    a.y = 1.0f;
  }

  // B operand: rows [w_pitch; w_beats; w_pos; summed bias]
  const int h = h0 + m;
  v2f bm;
  if (!hi) {
    bm.x = w_pitch[h];
    bm.y = w_beats[h];
  } else {
    bm.x = w_pos[h];
    bm.y = b_pitch[h] + b_beats[h] + b_pos[h];
  }

  // C operand: gathered encoder tile (L2-resident; enc is only 32MB vs 192MB L2)
  const int* idxb = idx + (size_t)b * Tn + t0;
  int rows[8];
#pragma unroll
  for (int r = 0; r < 8; ++r) rows[r] = idxb[r + 8 * hi];

  v8f c;
#pragma unroll
  for (int r = 0; r < 8; ++r)
    c[r] = enc[((size_t)b * Pn + rows[r]) * Hn + h];

  // D = A x B + C  (one matrix op does all three affine adds)
  v8f d = __builtin_amdgcn_wmma_f32_16x16x4_f32(
      /*neg_a=*/false, a, /*neg_b=*/false, bm,
      /*c_mod=*/(short)0, c, /*reuse_a=*/false, /*reuse_b=*/false);

  float* outb = out + ((size_t)b * Tn + t0) * Hn + h;
#pragma unroll
  for (int r = 0; r < 8; ++r)
    outb[(size_t)(r + 8 * hi) * Hn] = d[r];
}

// ---------------------------------------------------------------------------
extern "C" void kernel_launch(void* const* d_in, const int* in_sizes, int n_in,
                              void* d_out, int out_size, void* d_ws, size_t ws_size,
                              hipStream_t stream) {
  const float* enc     = (const float*)d_in[0];
  const float* pitch   = (const float*)d_in[1];
  const float* beats   = (const float*)d_in[2];
  const float* w_pitch = (const float*)d_in[3];
  const float* b_pitch = (const float*)d_in[4];
  const float* w_beats = (const float*)d_in[5];
  const float* b_beats = (const float*)d_in[6];
  const float* w_pos   = (const float*)d_in[7];
  const float* b_pos   = (const float*)d_in[8];
  const int*   align   = (const int*)d_in[9];
  float* out = (float*)d_out;
  int*   idx = (int*)d_ws;  // B*T int32 = 512 KB scratch

  align_scan_kernel<<<Bn, 1024, 0, stream>>>(align, idx);

  const int tiles = Bn * (Tn / 16) * (Hn / 128);  // 32768 blocks of 256 threads
  postnet_wmma_kernel<<<tiles, 256, 0, stream>>>(
      enc, pitch, beats, w_pitch, b_pitch, w_beats, b_beats, w_pos, b_pos,
      idx, out);
}